// SparsePolynomial_67190468379262
// MI455X (gfx1250) — compile-verified
//
#include <hip/hip_runtime.h>

#define D_DIM 4096
#define KEEP  2048   // max(1, int(4096 * 0.5))

// Native clang vector type: accepted by __builtin_nontemporal_* and lowers to
// 128-bit global_load_b128 / global_store_b128 with NT temporal hints.
typedef __attribute__((ext_vector_type(4))) float v4f;

// ---------------------------------------------------------------------------
// Kernel 1: exact top-k channel mask, matching jax.lax.top_k tie-breaking
// (rank = #{strictly greater} + #{equal with lower index}; keep if rank < KEEP)
// importance is staged in LDS so each block reads it from global once.
// ---------------------------------------------------------------------------
__global__ void topk_mask_kernel(const float* __restrict__ importance,
                                 float* __restrict__ mask) {
    __shared__ float s_imp[D_DIM];
    for (int i = threadIdx.x; i < D_DIM; i += blockDim.x) {
        s_imp[i] = importance[i];
    }
    __syncthreads();

    const int d = blockIdx.x * blockDim.x + threadIdx.x;
    if (d < D_DIM) {
        const float v = s_imp[d];
        int rank = 0;
        #pragma unroll 8
        for (int j = 0; j < D_DIM; ++j) {
            const float u = s_imp[j];
            rank += (u > v) ? 1 : ((u == v && j < d) ? 1 : 0);
        }
        mask[d] = (rank < KEEP) ? 1.0f : 0.0f;
    }
}

// ---------------------------------------------------------------------------
// Kernel 2: streaming elementwise polynomial with channel mask.
// out = mask ? ((c2*x + c1)*x + c0)*x : x
// 128-bit vectorized, non-temporal on the 256MB-in/256MB-out stream
// (working set > 192MB L2, so NT avoids cache thrash; mask stays hot via RT).
// ---------------------------------------------------------------------------
__global__ void __launch_bounds__(256)
sparse_poly_kernel(const v4f* __restrict__ x4,
                   const float* __restrict__ coeffs,
                   const float* __restrict__ mask,
                   v4f* __restrict__ out4,
                   int n4) {
    const int i = blockIdx.x * blockDim.x + threadIdx.x;
    if (i >= n4) return;

    // Uniform coefficient loads (small, cache-resident).
    const float c0 = coeffs[0];
    const float c1 = coeffs[1];
    const float c2 = coeffs[2];

    // Streamed 128-bit non-temporal load of x.
    const v4f xv = __builtin_nontemporal_load(&x4[i]);

    // Channel of this float4's first lane; D=4096 is a power of two, and
    // mbase is 16B-aligned since i*4 is a multiple of 4 floats.
    const int mbase = (i << 2) & (D_DIM - 1);
    const v4f mv = *reinterpret_cast<const v4f*>(&mask[mbase]);

    v4f ov;
    {
        const float t = xv.x;
        const float p = fmaf(fmaf(fmaf(c2, t, c1), t, c0), t, 0.0f);
        ov.x = (mv.x != 0.0f) ? p : t;
    }
    {
        const float t = xv.y;
        const float p = fmaf(fmaf(fmaf(c2, t, c1), t, c0), t, 0.0f);
        ov.y = (mv.y != 0.0f) ? p : t;
    }
    {
        const float t = xv.z;
        const float p = fmaf(fmaf(fmaf(c2, t, c1), t, c0), t, 0.0f);
        ov.z = (mv.z != 0.0f) ? p : t;
    }
    {
        const float t = xv.w;
        const float p = fmaf(fmaf(fmaf(c2, t, c1), t, c0), t, 0.0f);
        ov.w = (mv.w != 0.0f) ? p : t;
    }

    // Streamed 128-bit non-temporal store.
    __builtin_nontemporal_store(ov, &out4[i]);
}

// ---------------------------------------------------------------------------
// Launcher
//   d_in[0]: x          (4*4096*4096 fp32)
//   d_in[1]: coeffs     (3 fp32)
//   d_in[2]: importance (4096 fp32)
//   d_out  : y          (4*4096*4096 fp32)
//   d_ws   : mask       (4096 fp32) scratch
// ---------------------------------------------------------------------------
extern "C" void kernel_launch(void* const* d_in, const int* in_sizes, int n_in,
                              void* d_out, int out_size, void* d_ws, size_t ws_size,
                              hipStream_t stream) {
    const float* x          = (const float*)d_in[0];
    const float* coeffs     = (const float*)d_in[1];
    const float* importance = (const float*)d_in[2];
    float* out  = (float*)d_out;
    float* mask = (float*)d_ws;   // 4096 floats = 16 KB scratch

    // Build the top-k channel mask (16 blocks x 256 threads covers D=4096).
    topk_mask_kernel<<<D_DIM / 256, 256, 0, stream>>>(importance, mask);

    // Streaming polynomial pass: one float4 per thread.
    const int n  = in_sizes[0];          // 67,108,864 elements
    const int n4 = n >> 2;               // 16,777,216 float4s
    const int block = 256;
    const int grid  = (n4 + block - 1) / block;
    sparse_poly_kernel<<<grid, block, 0, stream>>>(
        (const v4f*)x, coeffs, mask, (v4f*)out, n4);
}